// SageHop_3015067042508
// MI455X (gfx1250) — compile-verified
//
#include <hip/hip_runtime.h>
#include <math.h>

#define HID 64
#define EDIMV 7
#define BNE 1e-3f

typedef __attribute__((ext_vector_type(2))) float v2f;
typedef __attribute__((ext_vector_type(8))) float v8f;

__device__ __forceinline__ void atom_addf(float* p, float v) { atomicAdd(p, v); }
__device__ __forceinline__ void atom_maxf(float* p, float v) {
  __hip_atomic_fetch_max(p, v, __ATOMIC_RELAXED, __HIP_MEMORY_SCOPE_AGENT);
}
__device__ __forceinline__ void atom_minf(float* p, float v) {
  __hip_atomic_fetch_min(p, v, __ATOMIC_RELAXED, __HIP_MEMORY_SCOPE_AGENT);
}
__device__ __forceinline__ float finz(float v) { return isfinite(v) ? v : 0.0f; }

// ----------------------------------------------------------------------------
// utility kernels
// ----------------------------------------------------------------------------
__global__ void k_fill(float* p, long n, float v) {
  long i = (long)blockIdx.x * blockDim.x + threadIdx.x;
  if (i < n) p[i] = v;
}

__global__ void k_copy_cols(const float* __restrict__ src, int lds,
                            float* __restrict__ dst, int ldd, long rows, int cols) {
  long i = (long)blockIdx.x * blockDim.x + threadIdx.x;
  if (i >= rows * cols) return;
  long r = i / cols; int c = (int)(i - r * cols);
  dst[r * ldd + c] = src[r * lds + c];
}

__global__ void k_edge_features(const float* __restrict__ x, const int* __restrict__ edges,
                                const float* __restrict__ bnb, const float* __restrict__ bng,
                                const float* __restrict__ bnm, const float* __restrict__ bnv,
                                float* __restrict__ e7, long E) {
  long e = (long)blockIdx.x * blockDim.x + threadIdx.x;
  if (e >= E) return;
  int s = edges[2 * e + 0], r = edges[2 * e + 1];
  float dx[6];
#pragma unroll
  for (int c = 0; c < 6; ++c) dx[c] = x[(long)r * 6 + c] - x[(long)s * 6 + c];
  float d = sqrtf(dx[0] * dx[0] + dx[1] * dx[1] + dx[2] * dx[2]);
  float inv = 1.0f / fmaxf(d, 1e-30f);
  float raw[EDIMV];
  raw[0] = dx[3]; raw[1] = dx[4]; raw[2] = dx[5]; raw[3] = d;
  raw[4] = d > 0.f ? dx[0] * inv : 0.f;
  raw[5] = d > 0.f ? dx[1] * inv : 0.f;
  raw[6] = d > 0.f ? dx[2] * inv : 0.f;
#pragma unroll
  for (int c = 0; c < EDIMV; ++c)
    e7[e * EDIMV + c] = (raw[c] - bnm[c]) * rsqrtf(bnv[c] + BNE) * bng[c] + bnb[c];
}

__global__ void k_degree(const int* __restrict__ edges, float* __restrict__ deg, long E) {
  long e = (long)blockIdx.x * blockDim.x + threadIdx.x;
  if (e < E) atomicAdd(&deg[edges[2 * e + 1]], 1.0f);
}

__global__ void k_gcount(const int* __restrict__ seg, float* __restrict__ g, long N) {
  long n = (long)blockIdx.x * blockDim.x + threadIdx.x;
  if (n < N) atomicAdd(&g[seg[n]], 1.0f);
}

__global__ void k_div_deg(float* __restrict__ buf, int ld, const float* __restrict__ deg,
                          long rows, int C) {
  long i = (long)blockIdx.x * blockDim.x + threadIdx.x;
  if (i >= rows * C) return;
  long r = i / C; int c = (int)(i - r * C);
  buf[r * ld + c] /= fmaxf(deg[r], 1.0f);
}

__global__ void k_fix_inf(float* __restrict__ buf, long n) {
  long i = (long)blockIdx.x * blockDim.x + threadIdx.x;
  if (i < n) buf[i] = finz(buf[i]);
}

// scatter-add 4 columns per thread (halves edge-index load redundancy)
__global__ void k_scatter_add4(const float* __restrict__ src, int lds,
                               const int* __restrict__ edges, float* __restrict__ dst,
                               int ldd, int C, long E) {
  const int CH = (C + 3) >> 2;
  long i = (long)blockIdx.x * blockDim.x + threadIdx.x;
  if (i >= E * CH) return;
  long e = i / CH;
  int c4 = (int)(i - e * CH) << 2;
  int s = edges[2 * e + 0], r = edges[2 * e + 1];
  const float* sp = src + (long)s * lds + c4;
  float* dp = dst + (long)r * ldd + c4;
  int rem = C - c4;
  atomicAdd(dp + 0, sp[0]);
  if (rem > 1) atomicAdd(dp + 1, sp[1]);
  if (rem > 2) atomicAdd(dp + 2, sp[2]);
  if (rem > 3) atomicAdd(dp + 3, sp[3]);
}

__global__ void k_bn(float* __restrict__ h, const float* __restrict__ gamma,
                     const float* __restrict__ beta, const float* __restrict__ mu,
                     const float* __restrict__ var, long n, int C) {
  long i = (long)blockIdx.x * blockDim.x + threadIdx.x;
  if (i >= n) return;
  int c = (int)(i % C);
  h[i] = (h[i] - mu[c]) * rsqrtf(var[c] + BNE) * gamma[c] + beta[c];
}

__global__ void k_rownorm_relu(float* __restrict__ z, long rows, int C) {
  int w = threadIdx.x >> 5, lane = threadIdx.x & 31;
  long row = (long)blockIdx.x * (blockDim.x >> 5) + w;
  if (row >= rows) return;
  float s = 0.f;
  for (int c = lane; c < C; c += 32) { float v = z[row * C + c]; s += v * v; }
  for (int off = 16; off; off >>= 1) s += __shfl_xor(s, off, 32);
  float rs = rsqrtf(fmaxf(s, 1e-12f));
  for (int c = lane; c < C; c += 32) {
    float v = z[row * C + c] * rs;
    z[row * C + c] = fmaxf(v, 0.f);
  }
}

// ---- graph pooling ----
__global__ void k_init_pooled(float* __restrict__ pooled, int B) {
  long i = (long)blockIdx.x * blockDim.x + threadIdx.x;
  if (i >= (long)B * 792) return;
  int c = (int)(i % 792);
  pooled[i] = (c < 256) ? -__builtin_inff() : 0.0f;
}

__global__ void k_pool_h(const float* __restrict__ h, const int* __restrict__ seg,
                         float* __restrict__ pooled, long N) {
  long i = (long)blockIdx.x * blockDim.x + threadIdx.x;
  if (i >= N * 256) return;
  long n = i / 256; int c = (int)(i - n * 256);
  float v = h[n * 256 + c];
  int b = seg[n];
  atom_maxf(&pooled[(long)b * 792 + c], v);
  atomicAdd(&pooled[(long)b * 792 + 512 + c], v);
}

__global__ void k_pool_x(const float* __restrict__ x, const int* __restrict__ seg,
                         float* __restrict__ xsum, float* __restrict__ xsq,
                         float* __restrict__ xmax, float* __restrict__ xmin, long N) {
  long i = (long)blockIdx.x * blockDim.x + threadIdx.x;
  if (i >= N * 6) return;
  long n = i / 6; int c = (int)(i - n * 6);
  float v = x[n * 6 + c];
  int b = seg[n];
  atomicAdd(&xsum[b * 6 + c], v);
  atomicAdd(&xsq[b * 6 + c], v * v);
  atom_maxf(&xmax[b * 6 + c], v);
  atom_minf(&xmin[b * 6 + c], v);
}

__global__ void k_final_hpool(float* __restrict__ pooled, const float* __restrict__ gcount, int B) {
  long i = (long)blockIdx.x * blockDim.x + threadIdx.x;
  if (i >= (long)B * 256) return;
  int b = (int)(i / 256), c = (int)(i % 256);
  float cnt = fmaxf(gcount[b], 1.0f);
  float* pb = pooled + (long)b * 792;
  pb[256 + c] = pb[512 + c] / cnt;
  pb[c] = finz(pb[c]);
}

__global__ void k_final_xglob(const float* __restrict__ xsum, const float* __restrict__ xsq,
                              const float* __restrict__ xmax, const float* __restrict__ xmin,
                              const float* __restrict__ gcount, float* __restrict__ pooled, int B) {
  long i = (long)blockIdx.x * blockDim.x + threadIdx.x;
  if (i >= (long)B * 6) return;
  int b = (int)(i / 6), c = (int)(i % 6);
  float cnt = fmaxf(gcount[b], 1.0f);
  float avg = xsum[i] / cnt;
  float var = fabsf(xsq[i] / cnt - avg * avg);
  float* pg = pooled + (long)b * 792 + 768;
  pg[c] = avg; pg[6 + c] = var; pg[12 + c] = finz(xmax[i]); pg[18 + c] = finz(xmin[i]);
}

__global__ void k_output(const float* __restrict__ xloge, const float* __restrict__ xang,
                         const float* __restrict__ xsig, const float* __restrict__ asW,
                         const float* __restrict__ asb, float* __restrict__ out, int B) {
  int b = blockIdx.x * blockDim.x + threadIdx.x;
  if (b >= B) return;
  float a0 = xang[b * 2 + 0], a1 = xang[b * 2 + 1];
  float z0 = a0 * asW[0] + a1 * asW[2] + asb[0];
  float z1 = a0 * asW[1] + a1 * asW[3] + asb[1];
  float zen = 1.0f / (1.0f + expf(-z0));
  float azi = 1.0f / (1.0f + expf(-z1));
  const float PI = 3.14159265358979323846f;
  out[b * 5 + 0] = xloge[b];
  out[b * 5 + 1] = zen * PI;
  out[b * 5 + 2] = azi * 2.0f * PI;
  out[b * 5 + 3] = fabsf(xsig[b * 2 + 0]) + 1e-5f;
  out[b * 5 + 4] = fabsf(xsig[b * 2 + 1]) + 1e-5f;
}

// ----------------------------------------------------------------------------
// Fused per-edge message MLP + scatter-aggregate. One wave = 16 edges.
// m=[x[recv],x[send],e] staged (zero-padded) in LDS; layer1 din->128 relu via
// V_WMMA_F32_16X16X4_F32 into LDS; layer2 128->64 relu via WMMA; atomic
// aggregate into out[recv]. MODE: 0=sum, 1=min, 2=max.
// ----------------------------------------------------------------------------
template <int XIN, int MODE>
__global__ __launch_bounds__(128) void k_edge_mlp(
    const float* __restrict__ xin, const float* __restrict__ e7, const int* __restrict__ edges,
    const float* __restrict__ W1, const float* __restrict__ b1,
    const float* __restrict__ W2, const float* __restrict__ b2,
    float* __restrict__ outbuf, long E) {
  constexpr int DIN = 2 * XIN + EDIMV;
  constexpr int K1 = (DIN + 3) & ~3;
  __shared__ float sm[4][16][K1];
  __shared__ float sh1[4][16][128];
  const int w = threadIdx.x >> 5;
  const int lane = threadIdx.x & 31;
  const long tileBase = (long)blockIdx.x * 64 + w * 16;

  if constexpr (XIN == 64) {
    // float4 staging: recv part (cols 0..63), send part (64..127), e part (128..135)
    for (int idx = lane; idx < 16 * 16; idx += 32) {
      int r = idx >> 4, c4 = idx & 15;
      long eid = tileBase + r;
      long eidc = (eid < E) ? eid : (E - 1);
      float m = (eid < E) ? 1.0f : 0.0f;
      float4 v = ((const float4*)xin)[(long)edges[2 * eidc + 1] * 16 + c4];
      v.x *= m; v.y *= m; v.z *= m; v.w *= m;
      *(float4*)&sm[w][r][c4 * 4] = v;
    }
    for (int idx = lane; idx < 16 * 16; idx += 32) {
      int r = idx >> 4, c4 = idx & 15;
      long eid = tileBase + r;
      long eidc = (eid < E) ? eid : (E - 1);
      float m = (eid < E) ? 1.0f : 0.0f;
      float4 v = ((const float4*)xin)[(long)edges[2 * eidc + 0] * 16 + c4];
      v.x *= m; v.y *= m; v.z *= m; v.w *= m;
      *(float4*)&sm[w][r][64 + c4 * 4] = v;
    }
    for (int idx = lane; idx < 16 * 8; idx += 32) {
      int r = idx >> 3, c = idx & 7;
      long eid = tileBase + r;
      long eidc = (eid < E) ? eid : (E - 1);
      int cc = (c < EDIMV) ? c : (EDIMV - 1);
      float v = e7[eidc * EDIMV + cc];
      if (eid >= E || c >= EDIMV) v = 0.0f;
      sm[w][r][128 + c] = v;
    }
  } else {
    for (int idx = lane; idx < 16 * K1; idx += 32) {
      int r = idx / K1, c = idx - r * K1;
      long eid = tileBase + r;
      long eidc = (eid < E) ? eid : (E - 1);
      float v;
      if (c < XIN)          v = xin[(long)edges[2 * eidc + 1] * XIN + c];
      else if (c < 2 * XIN) v = xin[(long)edges[2 * eidc + 0] * XIN + (c - XIN)];
      else {
        int cc = c - 2 * XIN;
        v = e7[eidc * EDIMV + ((cc < EDIMV) ? cc : (EDIMV - 1))];
        if (cc >= EDIMV) v = 0.0f;
      }
      if (eid >= E) v = 0.0f;
      sm[w][r][c] = v;
    }
  }

  const int mrow = lane & 15;            // A row (M) and C/D column (N)
  const int khalf = (lane >> 4) << 1;    // lanes 16-31 hold K+2,K+3
  const int rbase = (lane >> 4) << 3;    // C/D: lanes 16-31 hold rows M+8

  // layer 1: (16 x K1) @ (K1 x 128) + b1, relu -> sh1.
  // LDS A is zero-padded beyond DIN, so B index is clamped (product = 0).
  for (int ct = 0; ct < 8; ++ct) {
    float bias = b1[ct * 16 + mrow];
    v8f acc = {bias, bias, bias, bias, bias, bias, bias, bias};
#pragma unroll 2
    for (int k = 0; k < K1; k += 4) {
      int k0 = k + khalf;
      int kc0 = (k0 < DIN) ? k0 : (DIN - 1);
      int kc1 = (k0 + 1 < DIN) ? (k0 + 1) : (DIN - 1);
      v2f a, bb;
      a.x = sm[w][mrow][k0];
      a.y = sm[w][mrow][k0 + 1];
      bb.x = W1[(long)kc0 * 128 + ct * 16 + mrow];
      bb.y = W1[(long)kc1 * 128 + ct * 16 + mrow];
      acc = __builtin_amdgcn_wmma_f32_16x16x4_f32(false, a, false, bb, (short)0, acc, false, false);
    }
#pragma unroll
    for (int r = 0; r < 8; ++r)
      sh1[w][rbase + r][ct * 16 + mrow] = fmaxf(acc[r], 0.0f);
  }

  // layer 2: (16 x 128) @ (128 x 64) + b2, relu, atomic scatter-aggregate
  for (int ct = 0; ct < 4; ++ct) {
    float bias = b2[ct * 16 + mrow];
    v8f acc = {bias, bias, bias, bias, bias, bias, bias, bias};
#pragma unroll 2
    for (int k = 0; k < 128; k += 4) {
      int k0 = k + khalf;
      v2f a, bb;
      a.x = sh1[w][mrow][k0];
      a.y = sh1[w][mrow][k0 + 1];
      bb.x = W2[(long)k0 * HID + ct * 16 + mrow];
      bb.y = W2[(long)(k0 + 1) * HID + ct * 16 + mrow];
      acc = __builtin_amdgcn_wmma_f32_16x16x4_f32(false, a, false, bb, (short)0, acc, false, false);
    }
#pragma unroll
    for (int r = 0; r < 8; ++r) {
      long eid = tileBase + rbase + r;
      if (eid < E) {
        float v = fmaxf(acc[r], 0.0f);
        float* p = &outbuf[(long)edges[2 * eid + 1] * HID + ct * 16 + mrow];
        if (MODE == 0) atom_addf(p, v);
        else if (MODE == 1) atom_minf(p, v);
        else atom_maxf(p, v);
      }
    }
  }
}

// ----------------------------------------------------------------------------
// Fast WMMA GEMM: requires rows%16==0, K%4==0, dout%64==0.
// One wave computes a 16x64 tile (4 accumulators, A loaded once per K-step).
// C = act( concat(A1,A2) @ W + b ), W row-major (d1+d2) x dout.
// ----------------------------------------------------------------------------
__global__ __launch_bounds__(256) void k_gemm_fast(
    const float* __restrict__ A1, int ld1, int d1,
    const float* __restrict__ A2, int ld2, int d2,
    const float* __restrict__ W, const float* __restrict__ bvec,
    float* __restrict__ C, int ldc, long rows, int dout, int act) {
  const int w = threadIdx.x >> 5, lane = threadIdx.x & 31;
  const long rowTile = ((long)blockIdx.x * 8 + w) * 16;
  if (rowTile >= rows) return;  // wave-uniform
  const int colBase = blockIdx.y * 64;
  const int K = d1 + d2;
  const int mrow = lane & 15;
  const int khalf = (lane >> 4) << 1;
  const int rbase = (lane >> 4) << 3;
  const long arow = rowTile + mrow;
  const float* __restrict__ Ar1 = A1 + arow * ld1;
  const float* __restrict__ Ar2 = A2 ? (A2 + arow * ld2) : A1;
  const float* __restrict__ Wc = W + colBase + mrow;
  float bb0 = bvec ? bvec[colBase + mrow] : 0.0f;
  float bb1 = bvec ? bvec[colBase + 16 + mrow] : 0.0f;
  float bb2 = bvec ? bvec[colBase + 32 + mrow] : 0.0f;
  float bb3 = bvec ? bvec[colBase + 48 + mrow] : 0.0f;
  v8f acc0 = {bb0, bb0, bb0, bb0, bb0, bb0, bb0, bb0};
  v8f acc1 = {bb1, bb1, bb1, bb1, bb1, bb1, bb1, bb1};
  v8f acc2 = {bb2, bb2, bb2, bb2, bb2, bb2, bb2, bb2};
  v8f acc3 = {bb3, bb3, bb3, bb3, bb3, bb3, bb3, bb3};
#pragma unroll 2
  for (int k = 0; k < K; k += 4) {
    const int k0 = k + khalf;
    const float* pa0 = (k0 < d1) ? (Ar1 + k0) : (Ar2 + (k0 - d1));
    const float* pa1 = (k0 + 1 < d1) ? (Ar1 + k0 + 1) : (Ar2 + (k0 + 1 - d1));
    v2f a; a.x = *pa0; a.y = *pa1;
    const float* w0p = Wc + (long)k0 * dout;
    const float* w1p = Wc + (long)(k0 + 1) * dout;
    v2f b0; b0.x = w0p[0];  b0.y = w1p[0];
    v2f b1; b1.x = w0p[16]; b1.y = w1p[16];
    v2f b2; b2.x = w0p[32]; b2.y = w1p[32];
    v2f b3; b3.x = w0p[48]; b3.y = w1p[48];
    acc0 = __builtin_amdgcn_wmma_f32_16x16x4_f32(false, a, false, b0, (short)0, acc0, false, false);
    acc1 = __builtin_amdgcn_wmma_f32_16x16x4_f32(false, a, false, b1, (short)0, acc1, false, false);
    acc2 = __builtin_amdgcn_wmma_f32_16x16x4_f32(false, a, false, b2, (short)0, acc2, false, false);
    acc3 = __builtin_amdgcn_wmma_f32_16x16x4_f32(false, a, false, b3, (short)0, acc3, false, false);
  }
#pragma unroll
  for (int r = 0; r < 8; ++r) {
    long cr = rowTile + rbase + r;
    float v0 = acc0[r], v1 = acc1[r], v2 = acc2[r], v3 = acc3[r];
    if (act == 1) {
      v0 = fmaxf(v0, 0.f); v1 = fmaxf(v1, 0.f); v2 = fmaxf(v2, 0.f); v3 = fmaxf(v3, 0.f);
    } else if (act == 2) {
      v0 = v0 > 0.f ? v0 : 0.15f * v0; v1 = v1 > 0.f ? v1 : 0.15f * v1;
      v2 = v2 > 0.f ? v2 : 0.15f * v2; v3 = v3 > 0.f ? v3 : 0.15f * v3;
    }
    float* cp = C + cr * ldc + colBase + mrow;
    cp[0] = v0; cp[16] = v1; cp[32] = v2; cp[48] = v3;
  }
}

// ----------------------------------------------------------------------------
// General WMMA GEMM (used only for tiny head finals dout=1,2). Branchless:
// addresses are clamped in-bounds; out-of-range lanes just aren't stored.
// Requires K%4==0 (true for all calls) so no K-padding contamination.
// ----------------------------------------------------------------------------
__global__ __launch_bounds__(256) void k_gemm_gen(
    const float* __restrict__ A1, int ld1, int d1,
    const float* __restrict__ A2, int ld2, int d2,
    const float* __restrict__ W, const float* __restrict__ bvec,
    float* __restrict__ C, int ldc, long rows, int dout, int act) {
  const int w = threadIdx.x >> 5, lane = threadIdx.x & 31;
  const long rowTile = ((long)blockIdx.x * 8 + w) * 16;
  if (rowTile >= rows) return;
  const int colTile = blockIdx.y * 16;
  const int K = d1 + d2;
  const int mrow = lane & 15;
  const int khalf = (lane >> 4) << 1;
  const int rbase = (lane >> 4) << 3;
  long arow = rowTile + mrow;
  if (arow > rows - 1) arow = rows - 1;          // clamp (value unused if OOB)
  int col = colTile + mrow;
  const bool cok = col < dout;
  if (col > dout - 1) col = dout - 1;            // clamp
  const float* __restrict__ Ar1 = A1 + arow * ld1;
  const float* __restrict__ Ar2 = A2 ? (A2 + arow * ld2) : A1;
  float bias = bvec ? bvec[col] : 0.0f;
  v8f acc = {bias, bias, bias, bias, bias, bias, bias, bias};
  for (int k = 0; k < K; k += 4) {
    const int k0 = k + khalf;
    const float* pa0 = (k0 < d1) ? (Ar1 + k0) : (Ar2 + (k0 - d1));
    const float* pa1 = (k0 + 1 < d1) ? (Ar1 + k0 + 1) : (Ar2 + (k0 + 1 - d1));
    v2f a, bb;
    a.x = *pa0; a.y = *pa1;
    bb.x = W[(long)k0 * dout + col];
    bb.y = W[(long)(k0 + 1) * dout + col];
    acc = __builtin_amdgcn_wmma_f32_16x16x4_f32(false, a, false, bb, (short)0, acc, false, false);
  }
#pragma unroll
  for (int r = 0; r < 8; ++r) {
    long cr = rowTile + rbase + r;
    if (cr < rows && cok) {
      float v = acc[r];
      if (act == 1) v = fmaxf(v, 0.0f);
      else if (act == 2) v = v > 0.0f ? v : 0.15f * v;
      C[cr * ldc + (colTile + (lane & 15))] = v;
    }
  }
}

// ----------------------------------------------------------------------------
extern "C" void kernel_launch(void* const* d_in, const int* in_sizes, int n_in,
                              void* d_out, int out_size, void* d_ws, size_t ws_size,
                              hipStream_t stream) {
  (void)n_in; (void)out_size; (void)ws_size;
  const long N = in_sizes[0] / 6;
  const long E = in_sizes[1] / 2;
  const int B = 64;

  const float* x = (const float*)d_in[0];
  const int* edges = (const int*)d_in[1];
  const int* seg = (const int*)d_in[2];

  // params flattened in jax pytree order (dict keys sorted)
  const float* P[200];
  for (int i = 0; i < n_in && i < 200; ++i) P[i] = (const float*)d_in[i];
  const int I_ASW = 4, I_ASB = 5, I_ANG = 6, I_BN = 12, I_DEC = 16, I_GCN = 34;
  const int I_H12MAX1 = 38, I_H12MAX2 = 46, I_H12MIN1 = 58, I_H12MIN2 = 66;
  const int I_H23MIN2 = 78, I_H23MIN3 = 90, I_H2MEAN = 106, I_LOGE = 118, I_SIGS = 124;

  // ---- workspace carve ----
  float* wp = (float*)d_ws;
  auto alloc = [&](long n) { float* p = wp; wp += n; return p; };
  float* e7 = alloc(E * EDIMV);
  float* deg = alloc(N);
  float* gcount = alloc(B);
  float* na = alloc(N * HID);
  float* nb = alloc(N * HID);
  float* nc = alloc(N * HID);
  float* h = alloc(N * 262);
  float* hagg = alloc(N * 262);
  float* z1 = alloc(N * 128);
  float* z2 = alloc(N * 256);
  float* xsum = alloc(B * 6); float* xsq = alloc(B * 6);
  float* xmax = alloc(B * 6); float* xmin = alloc(B * 6);
  float* pooled = alloc((long)B * 792);
  float* dec0 = alloc((long)B * 2048);
  float* dec1 = alloc((long)B * 1024);
  float* dec2 = alloc((long)B * 512);
  float* t1 = alloc((long)B * 64);
  float* t2 = alloc((long)B * 64);
  float* xloge = alloc(B);
  float* xang = alloc(B * 2);
  float* xsig = alloc(B * 2);

  auto g1 = [](long n) { return dim3((unsigned)((n + 255) / 256)); };
  auto fill = [&](float* p, long n, float v) { k_fill<<<g1(n), 256, 0, stream>>>(p, n, v); };

  auto gemm = [&](const float* A1, int ld1, int d1, const float* A2, int ld2, int d2,
                  const float* W, const float* b, float* C, int ldc, long rows, int dout, int act) {
    const int K = d1 + d2;
    if (dout % 64 == 0 && K % 4 == 0 && rows % 16 == 0) {
      dim3 grid((unsigned)((rows + 127) / 128), (unsigned)(dout / 64));
      k_gemm_fast<<<grid, 256, 0, stream>>>(A1, ld1, d1, A2, ld2, d2, W, b, C, ldc, rows, dout, act);
    } else {
      dim3 grid((unsigned)((rows + 127) / 128), (unsigned)((dout + 15) / 16));
      k_gemm_gen<<<grid, 256, 0, stream>>>(A1, ld1, d1, A2, ld2, d2, W, b, C, ldc, rows, dout, act);
    }
  };

  auto edge_hop = [&](const float* xsrc, int xin, int mode, const float* W1, const float* b1,
                      const float* W2, const float* b2, float* dst) {
    const long n64 = N * HID;
    if (mode == 0) fill(dst, n64, 0.0f);
    else if (mode == 1) fill(dst, n64, INFINITY);
    else fill(dst, n64, -INFINITY);
    dim3 grid((unsigned)((E + 63) / 64));
    if (xin == 6) {
      if (mode == 0)      k_edge_mlp<6, 0><<<grid, 128, 0, stream>>>(xsrc, e7, edges, W1, b1, W2, b2, dst, E);
      else if (mode == 1) k_edge_mlp<6, 1><<<grid, 128, 0, stream>>>(xsrc, e7, edges, W1, b1, W2, b2, dst, E);
      else                k_edge_mlp<6, 2><<<grid, 128, 0, stream>>>(xsrc, e7, edges, W1, b1, W2, b2, dst, E);
    } else {
      if (mode == 0)      k_edge_mlp<64, 0><<<grid, 128, 0, stream>>>(xsrc, e7, edges, W1, b1, W2, b2, dst, E);
      else if (mode == 1) k_edge_mlp<64, 1><<<grid, 128, 0, stream>>>(xsrc, e7, edges, W1, b1, W2, b2, dst, E);
      else                k_edge_mlp<64, 2><<<grid, 128, 0, stream>>>(xsrc, e7, edges, W1, b1, W2, b2, dst, E);
    }
    if (mode == 0) k_div_deg<<<g1(n64), 256, 0, stream>>>(dst, HID, deg, N, HID);
    else k_fix_inf<<<g1(n64), 256, 0, stream>>>(dst, n64);
  };

  // one sgconv: K msg hops + 2-layer relu upd MLP, final into dst (ld=ldd)
  auto run_conv = [&](int base, int K, int xin, int mode, const float* x0, float* dst, int ldd) {
    const float* cur = x0; int curXin = xin;
    float* bufs[2] = {na, nb}; int t = 0;
    for (int i = 0; i < K; ++i) {
      float* o = bufs[t];
      edge_hop(cur, curXin, mode, P[base + 4 * i], P[base + 4 * i + 1],
               P[base + 4 * i + 2], P[base + 4 * i + 3], o);
      cur = o; curXin = HID; t ^= 1;
    }
    float* mid = bufs[t];
    gemm(cur, HID, HID, nullptr, 0, 0, P[base + 4 * K], P[base + 4 * K + 1], mid, HID, N, HID, 1);
    gemm(mid, HID, HID, nullptr, 0, 0, P[base + 4 * K + 2], P[base + 4 * K + 3], dst, ldd, N, HID, 1);
  };

  // ---- preliminaries ----
  fill(deg, N, 0.0f);
  k_degree<<<g1(E), 256, 0, stream>>>(edges, deg, E);
  fill(gcount, B, 0.0f);
  k_gcount<<<g1(N), 256, 0, stream>>>(seg, gcount, N);
  k_edge_features<<<g1(E), 256, 0, stream>>>(x, edges, P[I_BN + 0], P[I_BN + 1], P[I_BN + 2], P[I_BN + 3], e7, E);

  // ---- xglob pooling of raw x ----
  fill(xsum, B * 6, 0.0f); fill(xsq, B * 6, 0.0f);
  fill(xmax, B * 6, -INFINITY); fill(xmin, B * 6, INFINITY);
  k_pool_x<<<g1(N * 6), 256, 0, stream>>>(x, seg, xsum, xsq, xmax, xmin, N);

  // ---- conv stacks -> h = [x | x2me | x12mi | x12ma | x23mi]  (N x 262) ----
  k_copy_cols<<<g1(N * 6), 256, 0, stream>>>(x, 6, h, 262, N, 6);
  run_conv(I_H2MEAN, 2, 6, 0, x, h + 6, 262);
  run_conv(I_H12MIN1, 1, 6, 1, x, nc, HID);
  run_conv(I_H12MIN2, 2, 64, 1, nc, h + 70, 262);
  run_conv(I_H12MAX1, 1, 6, 2, x, nc, HID);
  run_conv(I_H12MAX2, 2, 64, 2, nc, h + 134, 262);
  run_conv(I_H23MIN2, 2, 6, 1, x, nc, HID);
  run_conv(I_H23MIN3, 3, 64, 1, nc, h + 198, 262);

  // ---- GCN layer 1 ----
  fill(hagg, N * 262, 0.0f);
  {
    int CH = (262 + 3) / 4;
    k_scatter_add4<<<g1(E * CH), 256, 0, stream>>>(h, 262, edges, hagg, 262, 262, E);
  }
  k_div_deg<<<g1(N * 262), 256, 0, stream>>>(hagg, 262, deg, N, 262);
  gemm(h, 262, 262, hagg, 262, 262, P[I_GCN + 0], P[I_GCN + 1], z1, 128, N, 128, 0);
  k_rownorm_relu<<<dim3((unsigned)((N + 7) / 8)), 256, 0, stream>>>(z1, N, 128);

  // ---- GCN layer 2 ----
  fill(hagg, N * 128, 0.0f);
  k_scatter_add4<<<g1(E * 32), 256, 0, stream>>>(z1, 128, edges, hagg, 128, 128, E);
  k_div_deg<<<g1(N * 128), 256, 0, stream>>>(hagg, 128, deg, N, 128);
  gemm(z1, 128, 128, hagg, 128, 128, P[I_GCN + 2], P[I_GCN + 3], z2, 256, N, 256, 0);
  k_rownorm_relu<<<dim3((unsigned)((N + 7) / 8)), 256, 0, stream>>>(z2, N, 256);

  // ---- graph pooling -> pooled (B x 792) = [max | mean | sum | xglob(24)] ----
  k_init_pooled<<<g1((long)B * 792), 256, 0, stream>>>(pooled, B);
  k_pool_h<<<g1(N * 256), 256, 0, stream>>>(z2, seg, pooled, N);
  k_final_hpool<<<g1((long)B * 256), 256, 0, stream>>>(pooled, gcount, B);
  k_final_xglob<<<g1((long)B * 6), 256, 0, stream>>>(xsum, xsq, xmax, xmin, gcount, pooled, B);

  // ---- decoder: leaky(lin) then BN, 792->2048->1024->512 ----
  gemm(pooled, 792, 792, nullptr, 0, 0, P[I_DEC + 0], P[I_DEC + 1], dec0, 2048, B, 2048, 2);
  k_bn<<<g1((long)B * 2048), 256, 0, stream>>>(dec0, P[I_DEC + 3], P[I_DEC + 2], P[I_DEC + 4], P[I_DEC + 5], (long)B * 2048, 2048);
  gemm(dec0, 2048, 2048, nullptr, 0, 0, P[I_DEC + 6], P[I_DEC + 7], dec1, 1024, B, 1024, 2);
  k_bn<<<g1((long)B * 1024), 256, 0, stream>>>(dec1, P[I_DEC + 9], P[I_DEC + 8], P[I_DEC + 10], P[I_DEC + 11], (long)B * 1024, 1024);
  gemm(dec1, 1024, 1024, nullptr, 0, 0, P[I_DEC + 12], P[I_DEC + 13], dec2, 512, B, 512, 2);
  k_bn<<<g1((long)B * 512), 256, 0, stream>>>(dec2, P[I_DEC + 15], P[I_DEC + 14], P[I_DEC + 16], P[I_DEC + 17], (long)B * 512, 512);

  // ---- heads (pure linear chains) ----
  gemm(dec2, 512, 512, nullptr, 0, 0, P[I_LOGE + 0], P[I_LOGE + 1], t1, 64, B, 64, 0);
  gemm(t1, 64, 64, nullptr, 0, 0, P[I_LOGE + 2], P[I_LOGE + 3], t2, 64, B, 64, 0);
  gemm(t2, 64, 64, nullptr, 0, 0, P[I_LOGE + 4], P[I_LOGE + 5], xloge, 1, B, 1, 0);

  gemm(dec2, 512, 512, nullptr, 0, 0, P[I_ANG + 0], P[I_ANG + 1], t1, 64, B, 64, 0);
  gemm(t1, 64, 64, nullptr, 0, 0, P[I_ANG + 2], P[I_ANG + 3], t2, 64, B, 64, 0);
  gemm(t2, 64, 64, nullptr, 0, 0, P[I_ANG + 4], P[I_ANG + 5], xang, 2, B, 2, 0);

  gemm(dec2, 512, 512, nullptr, 0, 0, P[I_SIGS + 0], P[I_SIGS + 1], t1, 64, B, 64, 0);
  gemm(t1, 64, 64, nullptr, 0, 0, P[I_SIGS + 2], P[I_SIGS + 3], t2, 64, B, 64, 0);
  gemm(t2, 64, 64, nullptr, 0, 0, P[I_SIGS + 4], P[I_SIGS + 5], xsig, 2, B, 2, 0);

  k_output<<<dim3(1), 64, 0, stream>>>(xloge, xang, xsig, P[I_ASW], P[I_ASB], (float*)d_out, B);
}